// Rope_MQA_25907242729623
// MI455X (gfx1250) — compile-verified
//
#include <hip/hip_runtime.h>
#include <math.h>
#include <stdint.h>

typedef float v2f __attribute__((ext_vector_type(2)));
typedef float v8f __attribute__((ext_vector_type(8)));

#define WMMA_F32(a, b, c) \
    __builtin_amdgcn_wmma_f32_16x16x4_f32(false, (a), false, (b), (short)0, (c), false, false)

// ---------------------------------------------------------------------------
// GEMM: C[M,N] = X[M,K] * op(W)
//   WT=true : W stored [N,K] row-major (compute X @ W^T)
//   WT=false: W stored [K,N] row-major (compute X @ W)
// One block (4 waves, 128 threads) computes a 64x64 C tile; wave wv owns the
// 16-row strip. W K-chunks (64x32 f32 = 8 KB) are staged cooperatively into
// LDS via CDNA5 async-to-LDS loads (ASYNCcnt), shared by all 4 waves. A
// fragments stream directly from global (per-wave rows). 32 WMMAs per wave
// per chunk against 8 global loads + 32 LDS loads.
// ---------------------------------------------------------------------------
template <bool WT>
__global__ __launch_bounds__(128) void gemm_f32(const float* __restrict__ X,
                                                const float* __restrict__ W,
                                                float* __restrict__ C,
                                                int M, int N, int K,
                                                int ldx, int ldw, int ldc) {
    __shared__ float wlds[64 * 32];  // one K-chunk of the W tile (8 KB)

    const int lane = threadIdx.x & 31;
    const int wv   = threadIdx.x >> 5;
    const int ntiles = N >> 6;
    const int mt = blockIdx.x / ntiles;  // 64-row block index
    const int nt = blockIdx.x % ntiles;  // 64-col block index

    const int n16  = lane & 15;
    const int hi   = lane >> 4;  // 0 or 1
    const int koff = hi << 1;    // 0 or 2

    const float* xrow = X + (size_t)(mt * 64 + wv * 16 + n16) * ldx;  // A row m=lane&15
    v8f acc[4] = {};

    for (int k0 = 0; k0 < K; k0 += 32) {
        __syncthreads();  // previous chunk fully consumed before restaging
        // ---- stage W chunk into LDS: 512 x 16B pieces, 4 per thread, async
#pragma unroll
        for (int i = 0; i < 4; ++i) {
            const int p = threadIdx.x * 4 + i;
            const float* g;
            const float* l;
            if (WT) {  // wlds[n][k], n-row = 32 floats, contiguous in global too
                const int n = p >> 3, kq = p & 7;
                g = W + (size_t)(nt * 64 + n) * ldw + k0 + kq * 4;
                l = &wlds[n * 32 + kq * 4];
            } else {   // wlds[k][n], k-row = 64 floats, contiguous in global too
                const int k = p >> 4, nq = p & 15;
                g = W + (size_t)(k0 + k) * ldw + nt * 64 + nq * 4;
                l = &wlds[k * 64 + nq * 4];
            }
            const uint32_t ldsa = (uint32_t)(uintptr_t)l;   // low 32 bits = LDS offset
            const uint64_t ga   = (uint64_t)(uintptr_t)g;
            asm volatile("global_load_async_to_lds_b128 %0, %1, off"
                         :: "v"(ldsa), "v"(ga) : "memory");
        }
        asm volatile("s_wait_asynccnt 0" ::: "memory");
        __syncthreads();  // all waves' staging visible

        // ---- 8 K-steps of 4 WMMAs
#pragma unroll
        for (int kk = 0; kk < 8; ++kk) {
            v2f a = *(const v2f*)(xrow + k0 + kk * 4 + koff);
#pragma unroll
            for (int t = 0; t < 4; ++t) {
                v2f b;
                if (WT) {
                    b = *(const v2f*)&wlds[(t * 16 + n16) * 32 + kk * 4 + koff];
                } else {
                    b.x = wlds[(kk * 4 + koff) * 64 + t * 16 + n16];
                    b.y = wlds[(kk * 4 + koff + 1) * 64 + t * 16 + n16];
                }
                acc[t] = WMMA_F32(a, b, acc[t]);
            }
        }
    }

#pragma unroll
    for (int t = 0; t < 4; ++t) {
#pragma unroll
        for (int v = 0; v < 8; ++v) {
            const int m = hi * 8 + v;  // C layout: vgpr v, lane group hi -> row v+8*hi
            C[(size_t)(mt * 64 + wv * 16 + m) * ldc + nt * 64 + t * 16 + n16] = acc[t][v];
        }
    }
}

// ---------------------------------------------------------------------------
// Split KV[M,256] into K (cols 0..127, pre-rope) and V (cols 128..255) outputs
// ---------------------------------------------------------------------------
__global__ __launch_bounds__(256) void copy_kv(const float* __restrict__ KV,
                                               float* __restrict__ outK,
                                               float* __restrict__ outV, int M) {
    const int idx = blockIdx.x * 256 + threadIdx.x;  // M*128 threads
    const int r = idx >> 7;
    const int d = idx & 127;
    outK[idx] = KV[(size_t)r * 256 + d];
    outV[idx] = KV[(size_t)r * 256 + 128 + d];
}

// ---------------------------------------------------------------------------
// RoPE on Q in place. One thread handles (d, d+64) pair of one (b,s,h).
// pos = past + s.  Threads: B*S*H*64
// ---------------------------------------------------------------------------
__global__ __launch_bounds__(256) void rope_q(float* __restrict__ Q,
                                              const int* __restrict__ past_ptr, int S) {
    const int idx = blockIdx.x * 256 + threadIdx.x;
    const int d = idx & 63;
    const int h = (idx >> 6) & 15;
    const int r = idx >> 10;  // b*S + s
    const int s = r % S;
    const float pos = (float)(*past_ptr + s);
    const float freq = __expf(-0.14391156213f * (float)d);  // 1/10000^(d/64)
    float sn, cs;
    sincosf(pos * freq, &sn, &cs);
    const size_t base = (size_t)r * 2048 + h * 128 + d;
    const float x1 = Q[base];
    const float x2 = Q[base + 64];
    Q[base] = x1 * cs - x2 * sn;
    Q[base + 64] = x2 * cs + x1 * sn;
}

// ---------------------------------------------------------------------------
// RoPE on K: read K half of KV[M,256], write roped K to KR[M,128]. pos = s.
// Threads: B*S*64
// ---------------------------------------------------------------------------
__global__ __launch_bounds__(256) void rope_k(const float* __restrict__ KV,
                                              float* __restrict__ KR, int S) {
    const int idx = blockIdx.x * 256 + threadIdx.x;
    const int d = idx & 63;
    const int r = idx >> 6;  // b*S + s
    const int s = r % S;
    const float freq = __expf(-0.14391156213f * (float)d);
    float sn, cs;
    sincosf((float)s * freq, &sn, &cs);
    const float x1 = KV[(size_t)r * 256 + d];
    const float x2 = KV[(size_t)r * 256 + 64 + d];
    KR[(size_t)r * 128 + d] = x1 * cs - x2 * sn;
    KR[(size_t)r * 128 + 64 + d] = x2 * cs + x1 * sn;
}

// ---------------------------------------------------------------------------
// Flash attention, causal, MQA (one KV head). One wave per (b, h, 16-row qtile).
// QK^T and P@V via fp32 WMMA. O written in place over the (roped) Q region of
// that exact tile/head -- safe: each wave stages its q tile into LDS first and
// tiles are disjoint across waves.
// ---------------------------------------------------------------------------
__global__ __launch_bounds__(128) void attn_kernel(float* __restrict__ QO,
                                                   const float* __restrict__ KR,
                                                   const float* __restrict__ KV,
                                                   const int* __restrict__ past_ptr,
                                                   int S) {
    __shared__ float qs[4][16 * 128];  // per-wave q tile, 8KB each
    __shared__ float ps[4][16 * 16];   // per-wave P tile (C-layout -> A-layout)

    const int lane = threadIdx.x & 31;
    const int wv   = threadIdx.x >> 5;
    const int qtiles = S >> 4;
    const int wid = blockIdx.x * 4 + wv;
    const int qt = wid % qtiles;
    const int h  = (wid / qtiles) & 15;
    const int b  = wid / (qtiles * 16);
    const int past = *past_ptr;

    const int n16 = lane & 15;
    const int hi  = lane >> 4;

    float* qsw = qs[wv];
    {
        float* qb = QO + (size_t)(b * S + qt * 16) * 2048 + h * 128;
        for (int i = lane; i < 16 * 128; i += 32)
            qsw[i] = qb[(size_t)(i >> 7) * 2048 + (i & 127)];
    }

    float mrow[8], lrow[8];
    v8f acc[8] = {};
#pragma unroll
    for (int v = 0; v < 8; ++v) { mrow[v] = -1e30f; lrow[v] = 0.f; }

    const float scale = 0.08838834764831845f;  // 1/sqrt(128)
    const int lastj = past + qt * 16 + 15;
    int nkt = lastj / 16 + 1;
    if (nkt > (S >> 4)) nkt = S >> 4;

    const float* krbase = KR + (size_t)b * S * 128;
    const float* vbase  = KV + (size_t)b * S * 256 + 128;  // V half
    float* psw = ps[wv];

    for (int kt = 0; kt < nkt; ++kt) {
        // ---- scores tile: 16x16 = q(16x128) @ k^T, 32 WMMAs of K=4
        v8f sc = {};
        const float* kp = krbase + (size_t)(kt * 16 + n16) * 128 + (hi << 1);
        const float* qp = qsw + n16 * 128 + (hi << 1);
#pragma unroll
        for (int kk = 0; kk < 32; ++kk) {
            v2f a  = *(const v2f*)(qp + kk * 4);
            v2f bb = *(const v2f*)(kp + kk * 4);
            sc = WMMA_F32(a, bb, sc);
        }

        // ---- scale + causal mask + online softmax (row = hi*8+v across 16 lanes)
        float pv[8];
        const int j = kt * 16 + n16;  // key position for this lane's column
#pragma unroll
        for (int v = 0; v < 8; ++v) {
            float x = sc[v] * scale;
            const int irow = past + qt * 16 + hi * 8 + v;
            if (j > irow) x = -1e30f;
            float t = x;
            t = fmaxf(t, __shfl_xor(t, 1));
            t = fmaxf(t, __shfl_xor(t, 2));
            t = fmaxf(t, __shfl_xor(t, 4));
            t = fmaxf(t, __shfl_xor(t, 8));
            const float mnew = fmaxf(mrow[v], t);
            const float p = __expf(x - mnew);
            float psum = p;
            psum += __shfl_xor(psum, 1);
            psum += __shfl_xor(psum, 2);
            psum += __shfl_xor(psum, 4);
            psum += __shfl_xor(psum, 8);
            const float rs = __expf(mrow[v] - mnew);
            lrow[v] = lrow[v] * rs + psum;
            mrow[v] = mnew;
            pv[v] = p;
#pragma unroll
            for (int cb = 0; cb < 8; ++cb) acc[cb][v] *= rs;
        }

        // ---- P tile: C-layout regs -> LDS -> A-layout fragments
#pragma unroll
        for (int v = 0; v < 8; ++v) psw[(hi * 8 + v) * 16 + n16] = pv[v];

        // ---- acc += P(16x16) @ V(16x128): 8 col-blocks x 4 WMMAs
        const float* pp  = psw + n16 * 16 + (hi << 1);
        const float* vp0 = vbase + (size_t)(kt * 16 + (hi << 1)) * 256 + n16;
#pragma unroll
        for (int cb = 0; cb < 8; ++cb) {
            const float* vp = vp0 + cb * 16;
            v8f c = acc[cb];
#pragma unroll
            for (int kk = 0; kk < 4; ++kk) {
                v2f a = *(const v2f*)(pp + kk * 4);
                v2f bb;
                bb.x = vp[(kk * 4 + 0) * 256];
                bb.y = vp[(kk * 4 + 1) * 256];
                c = WMMA_F32(a, bb, c);
            }
            acc[cb] = c;
        }
    }

    // ---- O = acc / l, written in place over this tile's q region
    float* ob = QO + (size_t)(b * S + qt * 16) * 2048 + h * 128;
#pragma unroll
    for (int cb = 0; cb < 8; ++cb) {
#pragma unroll
        for (int v = 0; v < 8; ++v) {
            const int m = hi * 8 + v;
            ob[(size_t)m * 2048 + cb * 16 + n16] = acc[cb][v] / lrow[v];
        }
    }
}

// ---------------------------------------------------------------------------
extern "C" void kernel_launch(void* const* d_in, const int* in_sizes, int n_in,
                              void* d_out, int out_size, void* d_ws, size_t ws_size,
                              hipStream_t stream) {
    const float* x   = (const float*)d_in[0];
    const float* wq  = (const float*)d_in[1];
    const float* wkv = (const float*)d_in[2];
    const float* wo  = (const float*)d_in[3];
    const int* past  = (const int*)d_in[4];

    const int B = 2, S = 2048, D = 2048, H = 16, DH = 128;
    const int M = B * S;  // 4096

    float* ws  = (float*)d_ws;
    float* Q   = ws;                        // M*D   (later holds O in place)
    float* KVt = Q + (size_t)M * D;         // M*256
    float* KR  = KVt + (size_t)M * 256;     // M*128
    // total ws use: (M*D + M*256 + M*128)*4 bytes = ~38 MB

    float* out  = (float*)d_out;            // M*D
    float* outK = out + (size_t)M * D;      // M*DH
    float* outV = outK + (size_t)M * DH;    // M*DH

    // 1) Q = x @ wq^T         [4096x2048] x [2048x2048]^T
    gemm_f32<true><<<(M / 64) * (D / 64), 128, 0, stream>>>(x, wq, Q, M, D, D, D, D, D);

    // 2) KV = x @ wkv         [4096x2048] x [2048x256]
    gemm_f32<false><<<(M / 64) * (256 / 64), 128, 0, stream>>>(x, wkv, KVt, M, 256, D, D, 256, 256);

    // 3) outputs K (pre-rope) and V
    copy_kv<<<(M * DH) / 256, 256, 0, stream>>>(KVt, outK, outV, M);

    // 4) RoPE
    rope_q<<<(M * H * 64) / 256, 256, 0, stream>>>(Q, past, S);
    rope_k<<<(M * 64) / 256, 256, 0, stream>>>(KVt, KR, S);

    // 5) causal MQA flash attention; O overwrites Q buffer
    attn_kernel<<<(B * H * (S / 16)) / 4, 128, 0, stream>>>(Q, KR, KVt, past, S);

    // 6) out = O @ wo^T
    gemm_f32<true><<<(M / 64) * (D / 64), 128, 0, stream>>>(Q, wo, out, M, D, D, D, D, D);
}